// LinkPredictor_87797721465322
// MI455X (gfx1250) — compile-verified
//
#include <hip/hip_runtime.h>

typedef __attribute__((ext_vector_type(2))) float v2f;
typedef __attribute__((ext_vector_type(4))) float v4f;
typedef __attribute__((ext_vector_type(8))) float v8f;

#define CC      0.01f
#define SC      0.1f
#define EPSC    1e-15f
#define ACLAMP  (1.0f - 1e-5f)

__device__ __forceinline__ float fast_rcp(float x)  { return __builtin_amdgcn_rcpf(x); }
__device__ __forceinline__ float fast_sqrt(float x) { return __builtin_amdgcn_sqrtf(x); }
__device__ __forceinline__ float silu(float s)      { return s * fast_rcp(1.0f + __expf(-s)); }

// atanh(clamp(z)) = 0.5*log((1+z)/(1-z))
__device__ __forceinline__ float atanh_clamped(float z) {
    z = fminf(fmaxf(z, -ACLAMP), ACLAMP);
    return 0.5f * __logf((1.0f + z) * fast_rcp(1.0f - z));
}

// ---------------------------------------------------------------------------
// Kernel 1: u = logmap0(x); sq = ||u||^2; d0 = 2/sqrt(c)*atanh(sqrt(c)*||u||)
// One wave32 per node (D=64 -> 2 elements per lane).
// ---------------------------------------------------------------------------
__global__ __launch_bounds__(128) void logmap_kernel(
    const float* __restrict__ x, float* __restrict__ u,
    float* __restrict__ sq, float* __restrict__ d0, int nNodes)
{
    int wave = (int)((blockIdx.x * blockDim.x + threadIdx.x) >> 5);
    int lane = threadIdx.x & 31;
    if (wave >= nNodes) return;

    v2f xv = *(const v2f*)(x + wave * 64 + lane * 2);
    float ss = xv.x * xv.x + xv.y * xv.y;
    #pragma unroll
    for (int off = 16; off > 0; off >>= 1) ss += __shfl_xor(ss, off, 32);

    float norm  = fast_sqrt(ss);
    float nc    = fmaxf(norm, EPSC);
    float scale = atanh_clamped(SC * norm) * fast_rcp(SC * nc);

    v2f uv; uv.x = xv.x * scale; uv.y = xv.y * scale;
    *(v2f*)(u + wave * 64 + lane * 2) = uv;

    if (lane == 0) {
        float un = scale * norm;            // ||u||
        sq[wave] = un * un;
        d0[wave] = (2.0f / SC) * atanh_clamped(SC * un);
    }
}

// ---------------------------------------------------------------------------
// Kernel 2: P[n,k] = u[n,:]@W1[0:64,k]   + d0[n]*W1[129,k]
//           Q[n,k] = u[n,:]@W1[64:128,k] + d0[n]*W1[130,k] + b1[k]
// ---------------------------------------------------------------------------
__global__ __launch_bounds__(256) void pq_kernel(
    const float* __restrict__ u, const float* __restrict__ d0,
    const float* __restrict__ W1, const float* __restrict__ b1,
    float* __restrict__ P, float* __restrict__ Q)
{
    int tid  = blockIdx.x * blockDim.x + threadIdx.x;
    int node = tid >> 6;
    int k    = tid & 63;
    const float* ur = u + node * 64;
    float dv = d0[node];
    float p = dv * W1[129 * 64 + k];
    float q = dv * W1[130 * 64 + k] + b1[k];
    #pragma unroll 8
    for (int d = 0; d < 64; ++d) {
        float ud = ur[d];
        p = fmaf(ud, W1[d * 64 + k], p);
        q = fmaf(ud, W1[(64 + d) * 64 + k], q);
    }
    P[tid] = p;
    Q[tid] = q;
}

// ---------------------------------------------------------------------------
// Kernel 3 (fused): per 16x16 (i,j) tile, one wave:
//   Gram tile g = U_i[16x64] * U_j^T[64x16] via V_WMMA_F32_16X16X4_F32 (x16),
//   then per pair: Poincare distance from (x2, y2, g) only, then
//   logit = sum_k W2[k]*silu(P[i,k]+Q[j,k]+d*wd[k]) + b2 ; out = sigmoid(logit)
// WMMA C layout: lane L, vgpr r -> (M = r + 8*(L>=16), N = L%16): each lane
// directly owns its 8 output pairs.
// ---------------------------------------------------------------------------
__global__ __launch_bounds__(128) void pair_kernel(
    const float* __restrict__ u, const float* __restrict__ sq,
    const float* __restrict__ P, const float* __restrict__ Q,
    const float* __restrict__ W1, const float* __restrict__ W2,
    const float* __restrict__ b2p, float* __restrict__ out, int N)
{
    __shared__ float Pt[4][16 * 68];   // stride 68 floats = 272B: 16B-aligned rows,
    __shared__ float Qt[4][16 * 68];   // conflict-free across 16 tj lanes
    __shared__ float wdl[64];
    __shared__ float w2l[64];

    const int w    = threadIdx.x >> 5;
    const int lane = threadIdx.x & 31;

    const int ntj  = N >> 4;           // j-tiles per row
    const int tpb  = ntj * ntj;        // tiles per batch
    int tile = blockIdx.x * 4 + w;
    int b    = tile / tpb;
    int t    = tile - b * tpb;
    int it   = t / ntj;
    int jt   = t - it * ntj;
    int i0   = it << 4;
    int j0   = jt << 4;

    if (threadIdx.x < 64)       wdl[threadIdx.x]      = W1[128 * 64 + threadIdx.x];
    else                        w2l[threadIdx.x - 64] = W2[threadIdx.x - 64];

    // Stage P rows (i-tile) and Q rows (j-tile) into LDS.
    const float* Pb = P + ((size_t)b * N + i0) * 64;
    const float* Qb = Q + ((size_t)b * N + j0) * 64;
    for (int idx = lane; idx < 16 * 64; idx += 32) {
        int r = idx >> 6, kc = idx & 63;
        Pt[w][r * 68 + kc] = Pb[idx];
        Qt[w][r * 68 + kc] = Qb[idx];
    }

    // --- Gram tile via f32 WMMA, K = 64 in 16 steps of 4 ---
    const float* Ub   = u + (size_t)b * N * 64;
    const int row     = lane & 15;
    const int khalf   = (lane >> 4) << 1;            // 0 or 2
    const float* Arow = Ub + (size_t)(i0 + row) * 64 + khalf;
    const float* Brow = Ub + (size_t)(j0 + row) * 64 + khalf;
    v8f c = {0.f, 0.f, 0.f, 0.f, 0.f, 0.f, 0.f, 0.f};
    #pragma unroll
    for (int kk = 0; kk < 16; ++kk) {
        v2f av = *(const v2f*)(Arow + kk * 4);
        v2f bv = *(const v2f*)(Brow + kk * 4);
        c = __builtin_amdgcn_wmma_f32_16x16x4_f32(false, av, false, bv,
                                                  (short)0, c, false, false);
    }

    __syncthreads();

    const int h  = lane >> 4;
    const int tj = lane & 15;
    const int jg = j0 + tj;
    const float y2  = sq[b * N + jg];
    const float b2v = b2p[0];
    const float* Prow0 = &Pt[w][(8 * h) * 68];
    const v4f* Qv  = (const v4f*)&Qt[w][tj * 68];
    const v4f* Wd  = (const v4f*)wdl;
    const v4f* W2v = (const v4f*)w2l;
    float* outb = out + (size_t)b * N * N;

    #pragma unroll
    for (int r = 0; r < 8; ++r) {
        int   ig = i0 + 8 * h + r;
        float g  = c[r];
        float x2 = sq[b * N + ig];

        // ||(-x) (+) y|| from (x2, y2, g) only
        float tc  = 1.0f - 2.0f * CC * g;
        float A   = tc + CC * y2;
        float Bc  = 1.0f - CC * x2;
        float den = fmaxf(tc + CC * CC * x2 * y2, EPSC);
        float m2  = A * A * x2 - 2.0f * A * Bc * g + Bc * Bc * y2;
        float z   = SC * fast_sqrt(fmaxf(m2, 0.0f)) * fast_rcp(den);
        float pd  = (2.0f / SC) * atanh_clamped(z);

        const v4f* Pv = (const v4f*)(Prow0 + r * 68);
        float acc = 0.0f;
        #pragma unroll
        for (int kv = 0; kv < 16; ++kv) {
            v4f s  = Pv[kv] + Qv[kv] + pd * Wd[kv];   // ds_load_b128 x3
            v4f wv = W2v[kv];
            acc = fmaf(silu(s.x), wv.x, acc);
            acc = fmaf(silu(s.y), wv.y, acc);
            acc = fmaf(silu(s.z), wv.z, acc);
            acc = fmaf(silu(s.w), wv.w, acc);
        }
        float logit = acc + b2v;
        outb[(size_t)ig * N + jg] = fast_rcp(1.0f + __expf(-logit));
    }
}

// ---------------------------------------------------------------------------
extern "C" void kernel_launch(void* const* d_in, const int* in_sizes, int n_in,
                              void* d_out, int out_size, void* d_ws, size_t ws_size,
                              hipStream_t stream) {
    const float* emb = (const float*)d_in[0];   // [B,N,64]
    const float* W1  = (const float*)d_in[1];   // [131,64]
    const float* b1  = (const float*)d_in[2];   // [64]
    const float* W2  = (const float*)d_in[3];   // [64,1]
    const float* b2  = (const float*)d_in[4];   // [1]
    float* out = (float*)d_out;                 // [B,N,N]

    const int BN = in_sizes[0] / 64;            // B*N = 1024 nodes
    const int N  = out_size / BN;               // 512
    // workspace layout (floats): u[BN*64] | sq[BN] | d0[BN] | P[BN*64] | Q[BN*64]
    float* u   = (float*)d_ws;
    float* sqv = u   + (size_t)BN * 64;
    float* d0v = sqv + BN;
    float* P   = d0v + BN;
    float* Q   = P   + (size_t)BN * 64;

    logmap_kernel<<<(BN + 3) / 4, 128, 0, stream>>>(emb, u, sqv, d0v, BN);
    pq_kernel<<<(BN * 64) / 256, 256, 0, stream>>>(u, d0v, W1, b1, P, Q);

    int tiles = out_size / 256;                 // B*(N/16)^2 = 2048
    pair_kernel<<<tiles / 4, 128, 0, stream>>>(u, sqv, P, Q, W1, W2, b2, out, N);
}